// GCN_11484742549902
// MI455X (gfx1250) — compile-verified
//
#include <hip/hip_runtime.h>
#include <hip/hip_bf16.h>

#define FD 128          // feature dim
#define BN_EPS 1e-5f

typedef __attribute__((ext_vector_type(2))) float v2f;
typedef __attribute__((ext_vector_type(8))) float v8f;

// ---------------------------------------------------------------------------
// Degree accumulation: deg_out[src]++, deg_in[dst]++  (float atomics into L2)
// ---------------------------------------------------------------------------
__global__ __launch_bounds__(256) void gcn_deg_kernel(const int* __restrict__ src,
                                                      const int* __restrict__ dst,
                                                      float* __restrict__ deg_out,
                                                      float* __restrict__ deg_in,
                                                      int E) {
  int e = blockIdx.x * 256 + threadIdx.x;
  if (e < E) {
    atomicAdd(&deg_out[src[e]], 1.0f);
    atomicAdd(&deg_in[dst[e]], 1.0f);
  }
}

// In-place: deg -> clamp(deg,1)^-1/2
__global__ __launch_bounds__(256) void gcn_norm_kernel(float* __restrict__ ns,
                                                       float* __restrict__ nd, int n) {
  int i = blockIdx.x * 256 + threadIdx.x;
  if (i < n) {
    ns[i] = rsqrtf(fmaxf(ns[i], 1.0f));
    nd[i] = rsqrtf(fmaxf(nd[i], 1.0f));
  }
}

// ---------------------------------------------------------------------------
// HS = (H * norm_src[:,None]) @ W   via V_WMMA_F32_16X16X4_F32
//   - workgroup = 256 threads = 8 waves; each wave: 32 rows x 128 cols
//     (two 16-row M-blocks -> every B fragment from LDS feeds 2 WMMAs,
//      halving LDS bytes per WMMA vs a single 16-row block)
//   - W (128x128 f32, 64KB) staged in LDS, pre-packed into B-fragment order:
//       VGPR0 = {K=kk   (lanes 0-15), K=kk+2 (lanes 16-31)}
//       VGPR1 = {K=kk+1 (lanes 0-15), K=kk+3 (lanes 16-31)}
//     packed offset = ((t*32 + kk)*32 + lane)*2 + v   (t = 16-col tile index)
//   - A fragments (ISA 16x4 f32 layout) software-pipelined one K-step ahead
//     and fused with norm_src scaling (scale=0 for pad rows -> EXEC stays
//     all-ones through the WMMAs; only the stores are lane-guarded).
// ---------------------------------------------------------------------------
__global__ __launch_bounds__(256) void gcn_gemm_wmma(const float* __restrict__ H,
                                                     const float* __restrict__ W,
                                                     const float* __restrict__ ns,
                                                     float* __restrict__ HS,
                                                     int n) {
  extern __shared__ float lds[];  // 128*128 floats = 64KB, packed B fragments
  const int tid = threadIdx.x;

  // Cooperative coalesced load of W, scattered pack into LDS.
  for (int idx = tid; idx < FD * FD; idx += 256) {
    const int r = idx >> 7;       // K index
    const int c = idx & 127;      // N (output col) index
    const float w = W[idx];
    const int t   = c >> 4;       // 16-col tile
    const int rem = r & 3;        // position within K-group of 4
    const int l   = (c & 15) | ((rem & 2) << 3);  // lane
    const int kk  = r >> 2;       // K-step
    lds[((((t << 5) | kk) << 5) | l) * 2 + (rem & 1)] = w;
  }
  __syncthreads();

  const int lane = tid & 31;
  const int wave = tid >> 5;
  const int row_base = (blockIdx.x * 8 + wave) * 32;  // 32 rows per wave
  if (row_base >= n) return;  // wave-uniform; no barriers after this point

  const int m = lane & 15;
  const int row0 = row_base + m;
  const int row1 = row_base + 16 + m;
  const bool inb0 = row0 < n;
  const bool inb1 = row1 < n;
  const float scale0 = inb0 ? ns[row0] : 0.0f;        // zero out pad rows
  const float scale1 = inb1 ? ns[row1] : 0.0f;
  const float* hrow0 = H + (long)(inb0 ? row0 : 0) * FD;
  const float* hrow1 = H + (long)(inb1 ? row1 : 0) * FD;
  const int khalf = (lane >> 4) << 1;                 // lanes 16-31 -> K+2

  v8f acc[16] = {};

  // Software-pipelined A fragments (one K-step ahead; wrap avoids OOB/branch).
  float2 av0 = *(const float2*)(hrow0 + khalf);
  float2 av1 = *(const float2*)(hrow1 + khalf);
  for (int kk = 0; kk < 32; ++kk) {
    const int koff = (((kk + 1) & 31) << 2) + khalf;
    const float2 av0n = *(const float2*)(hrow0 + koff);
    const float2 av1n = *(const float2*)(hrow1 + koff);
    v2f a0, a1;
    a0.x = av0.x * scale0;
    a0.y = av0.y * scale0;
    a1.x = av1.x * scale1;
    a1.y = av1.y * scale1;

    // Batch-load all 8 B fragments for this K-step; each feeds 2 WMMAs.
    const float* bp = lds + (kk << 6) + (lane << 1);
    v2f bf[8];
#pragma unroll
    for (int t = 0; t < 8; ++t) bf[t] = *(const v2f*)(bp + (t << 11));

#pragma unroll
    for (int t = 0; t < 8; ++t) {
      acc[t] = __builtin_amdgcn_wmma_f32_16x16x4_f32(
          false, a0, false, bf[t], (short)0, acc[t], false, false);
      acc[8 + t] = __builtin_amdgcn_wmma_f32_16x16x4_f32(
          false, a1, false, bf[t], (short)0, acc[8 + t], false, false);
    }
    av0 = av0n;
    av1 = av1n;
  }

  // C/D layout: lanes 0-15 -> M = v, N = lane; lanes 16-31 -> M = v+8.
  const int nc = lane & 15;
  const int mb = (lane >> 4) << 3;
#pragma unroll
  for (int blk = 0; blk < 2; ++blk) {
#pragma unroll
    for (int t = 0; t < 8; ++t) {
#pragma unroll
      for (int v = 0; v < 8; ++v) {
        const int r2 = row_base + blk * 16 + mb + v;
        if (r2 < n) HS[(long)r2 * FD + t * 16 + nc] = acc[blk * 8 + t][v];
      }
    }
  }
}

// ---------------------------------------------------------------------------
// Scatter-add SpMM: one wave per edge; lane handles 4 contiguous features.
// Coalesced 512B gather per wave + dword float atomics into agg (L2-resident).
// ---------------------------------------------------------------------------
__global__ __launch_bounds__(256) void gcn_scatter_kernel(const float* __restrict__ hs,
                                                          const int* __restrict__ src,
                                                          const int* __restrict__ dst,
                                                          float* __restrict__ agg,
                                                          int E) {
  const int gtid = blockIdx.x * 256 + threadIdx.x;
  const int e = gtid >> 5;
  const int lane = threadIdx.x & 31;
  if (e < E) {
    const int s = src[e];
    const int d = dst[e];
    const float4 v = *(const float4*)(hs + (long)s * FD + (lane << 2));
    float* ap = agg + (long)d * FD + (lane << 2);
    atomicAdd(ap + 0, v.x);
    atomicAdd(ap + 1, v.y);
    atomicAdd(ap + 2, v.z);
    atomicAdd(ap + 3, v.w);
  }
}

// ---------------------------------------------------------------------------
// Mid-layer epilogue: h = relu(BN(agg*norm_dst + b))   (float4 vectorized)
// ---------------------------------------------------------------------------
__global__ __launch_bounds__(256) void gcn_epi_bn_relu(const float* __restrict__ agg,
                                                       const float* __restrict__ ndv,
                                                       const float* __restrict__ bias,
                                                       const float* __restrict__ gamma,
                                                       const float* __restrict__ beta,
                                                       const float* __restrict__ mean,
                                                       const float* __restrict__ var,
                                                       float* __restrict__ hout, int n) {
  const int idx = blockIdx.x * 256 + threadIdx.x;
  const int i = idx >> 5;
  const int c = (idx & 31) << 2;
  if (i < n) {
    const float nd = ndv[i];
    const float4 v = *(const float4*)(agg + (long)i * FD + c);
    const float* vi = (const float*)&v;
    float4 o;
    float* vo = (float*)&o;
#pragma unroll
    for (int j = 0; j < 4; ++j) {
      float x = vi[j] * nd + bias[c + j];
      x = (x - mean[c + j]) * rsqrtf(var[c + j] + BN_EPS) * gamma[c + j] + beta[c + j];
      vo[j] = fmaxf(x, 0.0f);
    }
    *(float4*)(hout + (long)i * FD + c) = o;
  }
}

// Final epilogue: out = agg*norm_dst + b
__global__ __launch_bounds__(256) void gcn_epi_final(const float* __restrict__ agg,
                                                     const float* __restrict__ ndv,
                                                     const float* __restrict__ bias,
                                                     float* __restrict__ out, int n) {
  const int idx = blockIdx.x * 256 + threadIdx.x;
  const int i = idx >> 5;
  const int c = (idx & 31) << 2;
  if (i < n) {
    const float nd = ndv[i];
    const float4 v = *(const float4*)(agg + (long)i * FD + c);
    const float* vi = (const float*)&v;
    float4 o;
    float* vo = (float*)&o;
#pragma unroll
    for (int j = 0; j < 4; ++j) vo[j] = vi[j] * nd + bias[c + j];
    *(float4*)(out + (long)i * FD + c) = o;
  }
}

extern "C" void kernel_launch(void* const* d_in, const int* in_sizes, int n_in,
                              void* d_out, int out_size, void* d_ws, size_t ws_size,
                              hipStream_t stream) {
  const float* x   = (const float*)d_in[0];
  const int* esrc  = (const int*)d_in[1];
  const int* edst  = (const int*)d_in[2];
  const float* Wm[3] = {(const float*)d_in[3], (const float*)d_in[5], (const float*)d_in[7]};
  const float* bv[3] = {(const float*)d_in[4], (const float*)d_in[6], (const float*)d_in[8]};
  const float* gam[2] = {(const float*)d_in[9],  (const float*)d_in[13]};
  const float* bet[2] = {(const float*)d_in[10], (const float*)d_in[14]};
  const float* mea[2] = {(const float*)d_in[11], (const float*)d_in[15]};
  const float* var[2] = {(const float*)d_in[12], (const float*)d_in[16]};

  const int n = in_sizes[0] / FD;   // 100000
  const int E = in_sizes[1];        // 1000000

  // Workspace layout (floats): [ns:N][nd:N][hs:N*D][agg:N*D][h:N*D]
  float* ws   = (float*)d_ws;
  float* ns   = ws;
  float* ndv  = ws + n;
  float* hs   = ws + 2 * (long)n;
  float* agg  = hs + (long)n * FD;
  float* hbuf = agg + (long)n * FD;

  // Degrees -> norms
  hipMemsetAsync(ns, 0, 2 * (size_t)n * sizeof(float), stream);
  gcn_deg_kernel<<<(E + 255) / 256, 256, 0, stream>>>(esrc, edst, ns, ndv, E);
  gcn_norm_kernel<<<(n + 255) / 256, 256, 0, stream>>>(ns, ndv, n);

  const int gemm_blocks = (n + 255) / 256;              // 8 waves * 32 rows / WG
  const int scat_blocks = (int)(((long)E * 32 + 255) / 256);
  const int epi_blocks  = (int)(((long)n * 32 + 255) / 256);

  const float* hin = x;
  for (int layer = 0; layer < 3; ++layer) {
    gcn_gemm_wmma<<<gemm_blocks, 256, 64 * 1024, stream>>>(hin, Wm[layer], ns, hs, n);
    hipMemsetAsync(agg, 0, (size_t)n * FD * sizeof(float), stream);
    gcn_scatter_kernel<<<scat_blocks, 256, 0, stream>>>(hs, esrc, edst, agg, E);
    if (layer < 2) {
      gcn_epi_bn_relu<<<epi_blocks, 256, 0, stream>>>(agg, ndv, bv[layer], gam[layer],
                                                      bet[layer], mea[layer], var[layer],
                                                      hbuf, n);
      hin = hbuf;
    } else {
      gcn_epi_final<<<epi_blocks, 256, 0, stream>>>(agg, ndv, bv[layer], (float*)d_out, n);
    }
  }
}